// GraphConvSparse_60430189855386
// MI455X (gfx1250) — compile-verified
//
#include <hip/hip_runtime.h>
#include <hip/hip_bf16.h>
#include <math.h>

// Problem constants (match the reference generator)
#define BATCH 4
#define DIN   256
#define DOUT  64
#define NTILES (DOUT / 16)   // 4 N-tiles, register-blocked per wave

typedef __attribute__((ext_vector_type(2))) float v2f;
typedef __attribute__((ext_vector_type(8))) float v8f;

// ---------------------------------------------------------------------------
// Kernel 0: zero the accumulation buffer (d_out) — harness poisons it.
// ---------------------------------------------------------------------------
__global__ __launch_bounds__(256) void zero_f4_kernel(float4* __restrict__ p, int n4) {
    int i = blockIdx.x * blockDim.x + threadIdx.x;
    if (i < n4) {
        p[i] = make_float4(0.f, 0.f, 0.f, 0.f);
    }
}

// ---------------------------------------------------------------------------
// Kernel 1: dense GEMM  x[M,64] = inputs[M,256] @ weight[256,64]   (M = B*N)
// One wave32 per 16-row stripe, register-blocked over all 4 N-tiles:
//   - A fragment (16x4 f32) loaded ONCE per K-step (global_load_b64)
//   - 4 B fragments -> 4x V_WMMA_F32_16X16X4_F32 per K-step
// 64 K-steps of K=4. EXEC all-ones in every wave that executes WMMA
// (out-of-range waves exit before any WMMA; branch is wave-uniform).
// ---------------------------------------------------------------------------
__global__ __launch_bounds__(256) void gemm_wmma_f32_kernel(
        const float* __restrict__ inputs,   // [M, 256]
        const float* __restrict__ weight,   // [256, 64]
        float* __restrict__ x,              // [M, 64]
        int rowTiles)                       // ceil-covered by grid; M / 16
{
    const int lane = threadIdx.x & 31;
    const int wave = (blockIdx.x * blockDim.x + threadIdx.x) >> 5;
    const int tm   = wave;                 // row tile (16 rows of inputs)
    if (tm >= rowTiles) return;            // wave-uniform exit

    const int mRow  = tm * 16 + (lane & 15);   // A row this lane supplies
    const int nLane = lane & 15;               // B column within a 16-wide tile
    const int khalf = (lane >> 4) * 2;         // lanes 0-15: K=0,1  16-31: K=2,3

    const float* aPtr = inputs + (size_t)mRow * DIN + khalf;
    const float* bPtr = weight + (size_t)khalf * DOUT + nLane;

    v8f c0 = {}, c1 = {}, c2 = {}, c3 = {};

    #pragma unroll 4
    for (int k = 0; k < DIN; k += 4) {
        // A fragment: one aligned 8B load, shared by all 4 N-tiles
        v2f a = *(const v2f*)(aPtr + k);

        const float* bk0 = bPtr + (size_t)k * DOUT;        // row k   (or k+2)
        const float* bk1 = bPtr + (size_t)(k + 1) * DOUT;  // row k+1 (or k+3)

        v2f b0; b0.x = bk0[0];  b0.y = bk1[0];   // N-tile 0: cols nLane
        v2f b1; b1.x = bk0[16]; b1.y = bk1[16];  // N-tile 1: cols nLane+16
        v2f b2; b2.x = bk0[32]; b2.y = bk1[32];  // N-tile 2: cols nLane+32
        v2f b3; b3.x = bk0[48]; b3.y = bk1[48];  // N-tile 3: cols nLane+48

        c0 = __builtin_amdgcn_wmma_f32_16x16x4_f32(false, a, false, b0, (short)0, c0, false, false);
        c1 = __builtin_amdgcn_wmma_f32_16x16x4_f32(false, a, false, b1, (short)0, c1, false, false);
        c2 = __builtin_amdgcn_wmma_f32_16x16x4_f32(false, a, false, b2, (short)0, c2, false, false);
        c3 = __builtin_amdgcn_wmma_f32_16x16x4_f32(false, a, false, b3, (short)0, c3, false, false);
    }

    // C/D layout: VGPR v -> M = v (lanes 0-15) or v+8 (lanes 16-31); N = lane&15
    const int mBase = tm * 16 + (lane >> 4) * 8;
    float* xo = x + (size_t)mBase * DOUT + nLane;
    #pragma unroll
    for (int v = 0; v < 8; ++v) {
        float* row = xo + (size_t)v * DOUT;
        row[0]  = c0[v];
        row[16] = c1[v];
        row[32] = c2[v];
        row[48] = c3[v];
    }
}

// ---------------------------------------------------------------------------
// Kernel 2: edge-parallel scatter SpMM with f32 atomics.
//   out[b, row[e], d] += values[e] * x[b, col[e], d]    for b in 0..3
// Thread t -> (e = t>>6, d = t&63). x rows are L2-resident (10.2 MB << 192 MB).
// ---------------------------------------------------------------------------
__global__ __launch_bounds__(256) void spmm_atomic_kernel(
        const int*   __restrict__ rowIdx,   // [E]
        const int*   __restrict__ colIdx,   // [E]
        const float* __restrict__ vals,     // [E]
        const float* __restrict__ x,        // [B, N, 64]
        float*       __restrict__ out,      // [B, N, 64] (accumulator)
        int E, int N)
{
    const int g = blockIdx.x * blockDim.x + threadIdx.x;
    const int e = g >> 6;
    if (e >= E) return;
    const int d = g & 63;

    const int   r  = rowIdx[e];
    const int   ci = colIdx[e];
    const float v  = vals[e];

    #pragma unroll
    for (int b = 0; b < BATCH; ++b) {
        const float xv = x[((size_t)b * N + ci) * DOUT + d];
        atomicAdd(&out[((size_t)b * N + r) * DOUT + d], v * xv);
    }
}

// ---------------------------------------------------------------------------
// Kernel 3: in-place tanh on the accumulator.
// ---------------------------------------------------------------------------
__global__ __launch_bounds__(256) void tanh_kernel(float* __restrict__ p, int n) {
    int i = blockIdx.x * blockDim.x + threadIdx.x;
    if (i < n) {
        p[i] = tanhf(p[i]);
    }
}

// ---------------------------------------------------------------------------
// Launch: zero -> GEMM (WMMA) -> atomic SpMM -> tanh, all on `stream`.
// Inputs (setup_inputs order): indices[2,E] i32, values[E] f32,
//                              inputs[B,N,DIN] f32, weight[DIN,DOUT] f32.
// d_out: [B,N,DOUT] f32.  d_ws: x intermediate [B,N,DOUT] f32 (10.24 MB).
// ---------------------------------------------------------------------------
extern "C" void kernel_launch(void* const* d_in, const int* in_sizes, int n_in,
                              void* d_out, int out_size, void* d_ws, size_t ws_size,
                              hipStream_t stream) {
    const int*   indices = (const int*)  d_in[0];   // [2, E]
    const float* values  = (const float*)d_in[1];   // [E]
    const float* inputs  = (const float*)d_in[2];   // [B, N, DIN]
    const float* weight  = (const float*)d_in[3];   // [DIN, DOUT]

    const int E = in_sizes[1];
    const int N = in_sizes[2] / (BATCH * DIN);
    const int M = BATCH * N;                        // rows of the dense GEMM

    float* out = (float*)d_out;                     // [B, N, DOUT] accumulator
    float* x   = (float*)d_ws;                      // [B, N, DOUT] intermediate

    // 1) zero d_out (accumulator target for atomics)
    {
        const int n4 = out_size / 4;
        zero_f4_kernel<<<(n4 + 255) / 256, 256, 0, stream>>>((float4*)out, n4);
    }

    // 2) WMMA GEMM: one wave per 16-row stripe (all 4 N-tiles in registers)
    {
        const int rowTiles = M / 16;                      // 40000/16 = 2500
        const int blocks   = (rowTiles + 7) / 8;          // 8 waves / block
        gemm_wmma_f32_kernel<<<blocks, 256, 0, stream>>>(inputs, weight, x, rowTiles);
    }

    // 3) atomic scatter SpMM
    {
        const long long threads = (long long)E * DOUT;
        const int blocks = (int)((threads + 255) / 256);
        spmm_atomic_kernel<<<blocks, 256, 0, stream>>>(
            indices, indices + E, values, x, out, E, N);
    }

    // 4) tanh in place
    {
        tanh_kernel<<<(out_size + 255) / 256, 256, 0, stream>>>(out, out_size);
    }
}